// MetaNetLinearizedModel_57483842289850
// MI455X (gfx1250) — compile-verified
//
#include <hip/hip_runtime.h>

// ---------------------------------------------------------------------------
// MetaNet linearized model, fused on MI455X (gfx1250, wave32, WMMA f32).
//
// result[b] = feats[b] + du[b]@W2 + sum_t coefs[b,t]*(u[b]@dW2[t] + db2[t])
//   feats = gelu(x@W1+b1)@W2+b2
//   du    = gelu'(x@W1+b1) * ( sum_t coefs[b,t]*(x[b]@dW1[t] + db1[t]) )
//   coefs = relu(feats@mW1^T+mb1)@mW2^T+mb2
//
// Memory-bound: dW1 stream (302 MB) dominates -> fp32 V_WMMA_F32_16X16X4_F32.
// New this round: the K=12288 GEMMs stage their shared A operand (a 48 KB
// slice of x) into LDS once per block via gfx1250 async copies
// (global_load_async_to_lds_b128 + s_wait_asynccnt), cutting A re-read
// traffic 8x and freeing VMEM issue slots for the dW1 stream.
// ---------------------------------------------------------------------------

namespace {
constexpr int kB    = 16;
constexpr int kD    = 12288;
constexpr int kF    = 768;
constexpr int kHID  = 192;
constexpr int kT    = 8;
constexpr int kNT   = kF / 16;    // 48 N-tiles
constexpr int kKC   = 768;        // K-chunk for staged (K=12288) GEMMs
constexpr int kNKC  = kD / kKC;   // 16 chunks
constexpr int kWPB  = 8;          // waves per 256-thread block
constexpr int kLdsPad    = 4;              // row pad -> conflict-free b64 reads
constexpr int kLdsStride = kKC + kLdsPad;  // 772 floats
}  // namespace

typedef __attribute__((ext_vector_type(2))) float v2f;
typedef __attribute__((ext_vector_type(8))) float v8f;

// --- gfx1250 async global->LDS copy (ASYNCcnt) -----------------------------
__device__ __forceinline__ void async_copy_b128(const float* gp, float* lds_p) {
  auto lp = (__attribute__((address_space(3))) float*)lds_p;
  auto g1 = (const __attribute__((address_space(1))) float*)gp;
  asm volatile("global_load_async_to_lds_b128 %0, %1, off"
               :: "v"(lp), "v"(g1) : "memory");
}
__device__ __forceinline__ void wait_async_all() {
  asm volatile("s_wait_asynccnt 0x0" ::: "memory");
}

// One 16x16 output tile from global A / global B.
// f32 WMMA 16x16x4 layouts (ISA 7.12.2):
//   A: lane l -> m=l&15, vgpr v holds k = 2*(l>>4)+v  (contiguous pair -> b64)
//   B: lane l -> n=l&15, vgpr v holds k = 2*(l>>4)+v  (two b32, stride ldb)
//   C: vgpr r -> row r+8*(l>>4), col l&15
__device__ __forceinline__ v8f wmma_tile_f32(const float* __restrict__ A, int lda,
                                             const float* __restrict__ Bm, int ldb,
                                             int K) {
  const int lane = threadIdx.x & 31;
  const int n    = lane & 15;
  const int hi   = lane >> 4;
  v8f acc = {0.f, 0.f, 0.f, 0.f, 0.f, 0.f, 0.f, 0.f};
  const float* ap = A + (size_t)n * lda + 2 * hi;
  const float* bp = Bm + (size_t)(2 * hi) * ldb + n;
  for (int k = 0; k < K; k += 4) {
    v2f a = *reinterpret_cast<const v2f*>(ap);
    v2f b;
    b.x = bp[0];
    b.y = bp[ldb];
    acc = __builtin_amdgcn_wmma_f32_16x16x4_f32(false, a, false, b, (short)0, acc,
                                                false, false);
    ap += 4;
    bp += 4 * (size_t)ldb;
  }
  return acc;
}

// ---- staged GEMM: partials[kc][row][t][col] = x[:,kcK:+K] @ Bmat[t] -------
// Bmat = nmat matrices of [kD, kF] (nmat=1 -> W1, nmat=8 -> dW1).
// A slice (16 x 768) staged into LDS once per block via async copies; all 8
// waves of the block (same kc, different (t,nt) tiles) read it via ds_load.
__global__ void __launch_bounds__(256)
k_gemm_staged(const float* __restrict__ x, const float* __restrict__ Bbase,
              int nmat, float* __restrict__ outp) {
  __shared__ __align__(16) float As[16 * kLdsStride];  // 49.4 KB
  const int blocksPerKc = (nmat * kNT) / kWPB;         // 6 or 48
  const int kc    = blockIdx.x / blocksPerKc;
  const int tile0 = (blockIdx.x - kc * blocksPerKc) * kWPB;

  // ---- stage A: 16 rows x 768 cols, 16B per lane-op, 12 ops/thread -------
  constexpr int kChunks = (16 * kKC) / 4;        // 3072 b128 transfers
  constexpr int kCPerRow = kKC / 4;              // 192
  for (int c = threadIdx.x; c < kChunks; c += 256) {
    const int row = c / kCPerRow;
    const int col = (c - row * kCPerRow) * 4;
    async_copy_b128(x + (size_t)row * kD + (size_t)kc * kKC + col,
                    &As[row * kLdsStride + col]);
  }
  wait_async_all();
  __syncthreads();

  // ---- compute one 16x16 tile per wave -----------------------------------
  const int tile = tile0 + (threadIdx.x >> 5);
  const int t    = tile / kNT;
  const int nt   = tile - t * kNT;
  const int lane = threadIdx.x & 31;
  const int n    = lane & 15;
  const int hi   = lane >> 4;

  const float* Bm = Bbase + (size_t)t * kD * kF + (size_t)(kc * kKC) * kF + nt * 16;
  const float* bp = Bm + (size_t)(2 * hi) * kF + n;
  const int aoff  = n * kLdsStride + 2 * hi;

  v8f acc = {0.f, 0.f, 0.f, 0.f, 0.f, 0.f, 0.f, 0.f};
  for (int k = 0; k < kKC; k += 4) {
    v2f a = *reinterpret_cast<const v2f*>(&As[aoff + k]);  // ds_load_b64
    v2f b;
    b.x = bp[0];
    b.y = bp[kF];
    acc = __builtin_amdgcn_wmma_f32_16x16x4_f32(false, a, false, b, (short)0, acc,
                                                false, false);
    bp += 4 * (size_t)kF;
  }

  // outp[((kc*16 + row)*nmat + t)*kF + col]
  float* op = outp + ((size_t)kc * kB * nmat + t) * kF + nt * 16 + n;
#pragma unroll
  for (int r = 0; r < 8; ++r) op[(size_t)(r + 8 * hi) * nmat * kF] = acc[r];
}

// ---- activation: reduce h0 partials, add b1, tanh-gelu + derivative -------
__global__ void __launch_bounds__(256)
k_act(const float* __restrict__ h0p, const float* __restrict__ b1,
      float* __restrict__ u, float* __restrict__ gp) {
  const int i = blockIdx.x * 256 + threadIdx.x;  // [0, 12288)
  const int f = i % kF;
  float s = b1[f];
#pragma unroll
  for (int kc = 0; kc < kNKC; ++kc) s += h0p[(size_t)kc * kB * kF + i];
  const float c0 = 0.7978845608028654f, c1 = 0.044715f;
  const float xv = s, x2 = xv * xv;
  const float th = tanhf(c0 * (xv + c1 * xv * x2));
  u[i]  = 0.5f * xv * (1.f + th);
  gp[i] = 0.5f * (1.f + th) + 0.5f * xv * (1.f - th * th) * c0 * (1.f + 3.f * c1 * x2);
}

// ---- GEMM: feats = u @ W2 + b2 --------------------------------------------
__global__ void __launch_bounds__(256)
k_gemm_feats(const float* __restrict__ u, const float* __restrict__ W2,
             const float* __restrict__ b2, float* __restrict__ feats) {
  const int nt = blockIdx.x * kWPB + (threadIdx.x >> 5);  // [0, 48)
  v8f acc = wmma_tile_f32(u, kF, W2 + nt * 16, kF, kF);
  const int lane = threadIdx.x & 31;
  const int n    = lane & 15;
  const int hi   = lane >> 4;
  const int col  = nt * 16 + n;
  const float bias = b2[col];
#pragma unroll
  for (int r = 0; r < 8; ++r)
    feats[(size_t)(r + 8 * hi) * kF + col] = acc[r] + bias;
}

// ---- meta-net: coefs = relu(feats@mW1^T+mb1)@mW2^T+mb2 --------------------
__global__ void __launch_bounds__(256)
k_meta(const float* __restrict__ feats, const float* __restrict__ mW1,
       const float* __restrict__ mb1, const float* __restrict__ mW2,
       const float* __restrict__ mb2, float* __restrict__ coefs) {
  __shared__ float hh[kB * kHID];  // 12 KB
  for (int e = threadIdx.x; e < kB * kHID; e += 256) {
    const int b = e / kHID, j = e - b * kHID;
    const float* fr = feats + (size_t)b * kF;
    const float* wr = mW1 + (size_t)j * kF;
    float s = mb1[j];
    for (int f = 0; f < kF; ++f) s += fr[f] * wr[f];
    hh[e] = s > 0.f ? s : 0.f;
  }
  __syncthreads();
  if (threadIdx.x < kB * kT) {
    const int b = threadIdx.x / kT, t = threadIdx.x - b * kT;
    const float* hr = hh + b * kHID;
    const float* wr = mW2 + t * kHID;
    float s = mb2[t];
    for (int j = 0; j < kHID; ++j) s += hr[j] * wr[j];
    coefs[threadIdx.x] = s;
  }
}

// ---- du = gelu'(h0) * sum_t coefs[b,t]*(y1[b,t,:] + db1[t,:]) -------------
__global__ void __launch_bounds__(256)
k_du(const float* __restrict__ y1p, const float* __restrict__ db1,
     const float* __restrict__ coefs, const float* __restrict__ gp,
     float* __restrict__ du) {
  const int i = blockIdx.x * 256 + threadIdx.x;  // [0, 12288)
  const int b = i / kF, f = i - b * kF;
  float acc = 0.f;
#pragma unroll
  for (int t = 0; t < kT; ++t) {
    float s = db1[(size_t)t * kF + f];
#pragma unroll
    for (int kc = 0; kc < kNKC; ++kc)
      s += y1p[(((size_t)kc * kB + b) * kT + t) * kF + f];
    acc += coefs[b * kT + t] * s;
  }
  du[i] = gp[i] * acc;
}

// ---- GEMM: z = du @ W2 ----------------------------------------------------
__global__ void __launch_bounds__(256)
k_gemm_z(const float* __restrict__ du, const float* __restrict__ W2,
         float* __restrict__ z) {
  const int nt = blockIdx.x * kWPB + (threadIdx.x >> 5);  // [0, 48)
  v8f acc = wmma_tile_f32(du, kF, W2 + nt * 16, kF, kF);
  const int lane = threadIdx.x & 31;
  const int n    = lane & 15;
  const int hi   = lane >> 4;
#pragma unroll
  for (int r = 0; r < 8; ++r)
    z[(size_t)(r + 8 * hi) * kF + nt * 16 + n] = acc[r];
}

// ---- GEMM: y2[b][t][g] = u @ dW2[t] ---------------------------------------
__global__ void __launch_bounds__(256)
k_gemm_y2(const float* __restrict__ u, const float* __restrict__ dW2,
          float* __restrict__ y2) {
  const int wave = blockIdx.x * kWPB + (threadIdx.x >> 5);  // [0, 384)
  const int t    = wave / kNT;
  const int nt   = wave - t * kNT;
  const float* Bm = dW2 + (size_t)t * kF * kF + nt * 16;
  v8f acc = wmma_tile_f32(u, kF, Bm, kF, kF);
  const int lane = threadIdx.x & 31;
  const int n    = lane & 15;
  const int hi   = lane >> 4;
  float* out = y2 + (size_t)t * kF + nt * 16 + n;  // (row*kT + t)*kF + col
#pragma unroll
  for (int r = 0; r < 8; ++r) out[(size_t)(r + 8 * hi) * (kT * kF)] = acc[r];
}

// ---- final: out = feats + z + sum_t coefs*(y2 + db2) ----------------------
__global__ void __launch_bounds__(256)
k_final(const float* __restrict__ feats, const float* __restrict__ z,
        const float* __restrict__ y2, const float* __restrict__ db2,
        const float* __restrict__ coefs, float* __restrict__ out) {
  const int i = blockIdx.x * 256 + threadIdx.x;  // [0, 12288)
  const int b = i / kF, f = i - b * kF;
  float s = feats[i] + z[i];
#pragma unroll
  for (int t = 0; t < kT; ++t)
    s += coefs[b * kT + t] * (y2[((size_t)b * kT + t) * kF + f] + db2[(size_t)t * kF + f]);
  out[i] = s;
}

extern "C" void kernel_launch(void* const* d_in, const int* in_sizes, int n_in,
                              void* d_out, int out_size, void* d_ws, size_t ws_size,
                              hipStream_t stream) {
  (void)in_sizes; (void)n_in; (void)out_size; (void)ws_size;
  const float* x   = (const float*)d_in[0];
  const float* W1  = (const float*)d_in[1];
  const float* b1  = (const float*)d_in[2];
  const float* W2  = (const float*)d_in[3];
  const float* b2  = (const float*)d_in[4];
  const float* mW1 = (const float*)d_in[5];
  const float* mb1 = (const float*)d_in[6];
  const float* mW2 = (const float*)d_in[7];
  const float* mb2 = (const float*)d_in[8];
  const float* dW1 = (const float*)d_in[9];
  const float* db1 = (const float*)d_in[10];
  const float* dW2 = (const float*)d_in[11];
  const float* db2 = (const float*)d_in[12];
  float* out = (float*)d_out;

  // workspace carve-up (floats); total ~1.93M floats (~7.7 MB)
  float* ws    = (float*)d_ws;
  float* h0p   = ws;                                 // kNKC*kB*kF      = 196608
  float* y1p   = h0p + (size_t)kNKC * kB * kF;       // kNKC*kB*kT*kF   = 1572864
  float* u     = y1p + (size_t)kNKC * kB * kT * kF;  // 12288
  float* gp    = u + kB * kF;
  float* feats = gp + kB * kF;
  float* du    = feats + kB * kF;
  float* z     = du + kB * kF;
  float* y2    = z + kB * kF;                        // kB*kT*kF = 98304
  float* coefs = y2 + (size_t)kB * kT * kF;          // 128

  // big dW1 stream first (6144 tile-waves), then the dependent chain
  k_gemm_staged<<<kNKC * (kT * kNT / kWPB), 256, 0, stream>>>(x, dW1, kT, y1p);
  k_gemm_staged<<<kNKC * (kNT / kWPB), 256, 0, stream>>>(x, W1, 1, h0p);
  k_act<<<(kB * kF) / 256, 256, 0, stream>>>(h0p, b1, u, gp);
  k_gemm_feats<<<kNT / kWPB, 256, 0, stream>>>(u, W2, b2, feats);
  k_gemm_y2<<<(kT * kNT) / kWPB, 256, 0, stream>>>(u, dW2, y2);
  k_meta<<<1, 256, 0, stream>>>(feats, mW1, mb1, mW2, mb2, coefs);
  k_du<<<(kB * kF) / 256, 256, 0, stream>>>(y1p, db1, coefs, gp, du);
  k_gemm_z<<<kNT / kWPB, 256, 0, stream>>>(du, W2, z);
  k_final<<<(kB * kF) / 256, 256, 0, stream>>>(feats, z, y2, db2, coefs, out);
}